// TransactionGNN_30167850287287
// MI455X (gfx1250) — compile-verified
//
#include <hip/hip_runtime.h>
#include <math.h>

typedef float v2f __attribute__((ext_vector_type(2)));
typedef float v8f __attribute__((ext_vector_type(8)));

// ---------------------------------------------------------------------------
// Small elementwise helpers
// ---------------------------------------------------------------------------
__global__ void k_fill(float* __restrict__ p, float v, int n) {
    int i = blockIdx.x * blockDim.x + threadIdx.x;
    if (i < n) p[i] = v;
}

__global__ void k_zero(float* __restrict__ p, int n) {
    int i = blockIdx.x * blockDim.x + threadIdx.x;
    if (i < n) p[i] = 0.0f;
}

// deg[dst] += 1 for each edge (deg pre-initialized to 1.0 for self-loop)
__global__ void k_degree(const int* __restrict__ dst, float* __restrict__ deg, int nedges) {
    int e = blockIdx.x * blockDim.x + threadIdx.x;
    if (e < nedges) atomicAdd(&deg[dst[e]], 1.0f);
}

__global__ void k_rsqrt_inplace(float* __restrict__ p, int n) {
    int i = blockIdx.x * blockDim.x + threadIdx.x;
    if (i < n) p[i] = rsqrtf(p[i]);
}

// ---------------------------------------------------------------------------
// Dense GEMM via V_WMMA_F32_16X16X4_F32  (fp32, wave32)
//   Y[N x FOUT] = X[N x FIN] @ W[FIN x FOUT]
// Block = 256 threads = 8 waves; each wave computes a 16-row stripe x FOUT.
// A frag (16x4 f32): lanes 0-15 hold row M=lane, K = k,k+1 in (v0,v1);
//                    lanes 16-31 hold row M=lane-16, K = k+2,k+3.
// B frag (4x16 f32): mirrored K striping, N = lane&15 within each VGPR.
// C/D (16x16 f32, 8 VGPRs): VGPR v -> M = v + 8*(lane>=16), N = lane&15.
// ---------------------------------------------------------------------------
template <int FIN, int FOUT>
__global__ void k_gemm_wmma(const float* __restrict__ X, const float* __restrict__ W,
                            float* __restrict__ Y, int nrows) {
    __shared__ float sW[FIN * FOUT];
    for (int i = threadIdx.x; i < FIN * FOUT; i += blockDim.x) sW[i] = W[i];
    __syncthreads();

    const int wave  = threadIdx.x >> 5;
    const int lane  = threadIdx.x & 31;
    const int row0  = (blockIdx.x * 8 + wave) * 16;
    if (row0 >= nrows) return;

    const int lhalf = lane >> 4;   // 0 or 1
    const int lmod  = lane & 15;

    constexpr int NT = FOUT / 16;
    v8f acc[NT] = {};

    const float* xrow = X + (size_t)(row0 + lmod) * FIN;

#pragma unroll
    for (int k = 0; k < FIN; k += 4) {
        // A fragment: 8-byte contiguous load per lane
        v2f a = *(const v2f*)(xrow + k + 2 * lhalf);
#pragma unroll
        for (int t = 0; t < NT; ++t) {
            const int n0 = t * 16;
            v2f b;
            b.x = sW[(k + 2 * lhalf)     * FOUT + n0 + lmod];
            b.y = sW[(k + 2 * lhalf + 1) * FOUT + n0 + lmod];
            acc[t] = __builtin_amdgcn_wmma_f32_16x16x4_f32(
                false, a, false, b, (short)0, acc[t], false, false);
        }
    }

#pragma unroll
    for (int t = 0; t < NT; ++t) {
        const int n0 = t * 16;
#pragma unroll
        for (int v = 0; v < 8; ++v) {
            const int m = v + 8 * lhalf;
            Y[(size_t)(row0 + m) * FOUT + n0 + lmod] = acc[t][v];
        }
    }
}

// ---------------------------------------------------------------------------
// Edge scatter-aggregate: OUT[dst] += XW[src] * dinv[src]*dinv[dst]
// One wave per edge; lanes cover the feature row (coalesced gather + atomics).
// ---------------------------------------------------------------------------
template <int F>
__global__ void k_scatter(const float* __restrict__ XW,
                          const int* __restrict__ esrc, const int* __restrict__ edst,
                          const float* __restrict__ dinv,
                          float* __restrict__ OUT, int nedges) {
    const int gwarp = (blockIdx.x * blockDim.x + threadIdx.x) >> 5;
    const int lane  = threadIdx.x & 31;
    if (gwarp >= nedges) return;

    const int s = esrc[gwarp];
    const int d = edst[gwarp];
    const float norm = dinv[s] * dinv[d];

    const float* src_row = XW + (size_t)s * F;
    float* dst_row = OUT + (size_t)d * F;

#pragma unroll
    for (int f = lane; f < F; f += 32)
        atomicAdd(dst_row + f, src_row[f] * norm);
}

// F = 2 final layer: one thread per edge
__global__ void k_scatter_f2(const float* __restrict__ XW,
                             const int* __restrict__ esrc, const int* __restrict__ edst,
                             const float* __restrict__ dinv,
                             float* __restrict__ OUT, int nedges) {
    const int e = blockIdx.x * blockDim.x + threadIdx.x;
    if (e >= nedges) return;
    const int s = esrc[e];
    const int d = edst[e];
    const float norm = dinv[s] * dinv[d];
    atomicAdd(&OUT[2 * d + 0], XW[2 * s + 0] * norm);
    atomicAdd(&OUT[2 * d + 1], XW[2 * s + 1] * norm);
}

// ---------------------------------------------------------------------------
// Epilogue: H = relu(AGG + XW*dinv^2 + bias)   (self-loop folded in; in place)
// ---------------------------------------------------------------------------
template <int F>
__global__ void k_finalize_relu(float* __restrict__ AGG_H, const float* __restrict__ XW,
                                const float* __restrict__ dinv, const float* __restrict__ bias,
                                int n) {
    const int i = blockIdx.x * blockDim.x + threadIdx.x;
    if (i >= n * F) return;
    const int node = i / F;
    const int f    = i % F;
    const float di = dinv[node];
    const float v  = AGG_H[i] + XW[i] * di * di + bias[f];
    AGG_H[i] = fmaxf(v, 0.0f);
}

// Layer 3 GEMM (32 -> 2): tiny, plain VALU
__global__ void k_gemm3(const float* __restrict__ H2, const float* __restrict__ W3,
                        float* __restrict__ XW3, int n) {
    const int i = blockIdx.x * blockDim.x + threadIdx.x;
    if (i >= n) return;
    const float* h = H2 + (size_t)i * 32;
    float a0 = 0.0f, a1 = 0.0f;
#pragma unroll
    for (int k = 0; k < 32; ++k) {
        const float hv = h[k];
        a0 += hv * W3[2 * k + 0];
        a1 += hv * W3[2 * k + 1];
    }
    XW3[2 * i + 0] = a0;
    XW3[2 * i + 1] = a1;
}

// out = log_softmax(AGG3 + XW3*dinv^2 + b3) over 2 classes
__global__ void k_logsoftmax2(const float* __restrict__ AGG3, const float* __restrict__ XW3,
                              const float* __restrict__ dinv, const float* __restrict__ b3,
                              float* __restrict__ out, int n) {
    const int i = blockIdx.x * blockDim.x + threadIdx.x;
    if (i >= n) return;
    const float di = dinv[i];
    const float z0 = AGG3[2 * i + 0] + XW3[2 * i + 0] * di * di + b3[0];
    const float z1 = AGG3[2 * i + 1] + XW3[2 * i + 1] * di * di + b3[1];
    const float m  = fmaxf(z0, z1);
    const float lse = m + logf(__expf(z0 - m) + __expf(z1 - m));
    out[2 * i + 0] = z0 - lse;
    out[2 * i + 1] = z1 - lse;
}

// ---------------------------------------------------------------------------
// Host-side orchestration
// ---------------------------------------------------------------------------
static inline int cdiv(long long a, long long b) { return (int)((a + b - 1) / b); }

extern "C" void kernel_launch(void* const* d_in, const int* in_sizes, int n_in,
                              void* d_out, int out_size, void* d_ws, size_t ws_size,
                              hipStream_t stream) {
    const float* x  = (const float*)d_in[0];
    const int*   ei = (const int*)d_in[1];   // [2, E] row-major int32
    const float* W1 = (const float*)d_in[2];
    const float* b1 = (const float*)d_in[3];
    const float* W2 = (const float*)d_in[4];
    const float* b2 = (const float*)d_in[5];
    const float* W3 = (const float*)d_in[6];
    const float* b3 = (const float*)d_in[7];

    const int N = in_sizes[0] / 128;   // 100000
    const int E = in_sizes[1] / 2;     // 3200000
    const int* esrc = ei;
    const int* edst = ei + E;

    // Workspace layout
    char* ws = (char*)d_ws;
    float* dinv = (float*)ws;                                   // N floats
    size_t offA = ((size_t)N * 4 + 255) & ~(size_t)255;
    float* bufA = (float*)(ws + offA);                          // N*64 floats (xw1 / xw2 / xw3+agg3)
    size_t offB = offA + (((size_t)N * 64 * 4 + 255) & ~(size_t)255);
    float* bufB = (float*)(ws + offB);                          // N*64 floats (agg/h)
    float* xw3  = bufA;
    float* agg3 = bufA + (size_t)N * 2;

    const int TB = 256;

    // --- degree / dinv ---
    k_fill<<<cdiv(N, TB), TB, 0, stream>>>(dinv, 1.0f, N);      // self-loop
    k_degree<<<cdiv(E, TB), TB, 0, stream>>>(edst, dinv, E);
    k_rsqrt_inplace<<<cdiv(N, TB), TB, 0, stream>>>(dinv, N);

    // --- layer 1: 128 -> 64 ---
    k_gemm_wmma<128, 64><<<cdiv(N, 128), TB, 0, stream>>>(x, W1, bufA, N);
    k_zero<<<cdiv((long long)N * 64, TB), TB, 0, stream>>>(bufB, N * 64);
    k_scatter<64><<<cdiv((long long)E * 32, TB), TB, 0, stream>>>(bufA, esrc, edst, dinv, bufB, E);
    k_finalize_relu<64><<<cdiv((long long)N * 64, TB), TB, 0, stream>>>(bufB, bufA, dinv, b1, N);

    // --- layer 2: 64 -> 32 ---
    k_gemm_wmma<64, 32><<<cdiv(N, 128), TB, 0, stream>>>(bufB, W2, bufA, N);
    k_zero<<<cdiv((long long)N * 32, TB), TB, 0, stream>>>(bufB, N * 32);
    k_scatter<32><<<cdiv((long long)E * 32, TB), TB, 0, stream>>>(bufA, esrc, edst, dinv, bufB, E);
    k_finalize_relu<32><<<cdiv((long long)N * 32, TB), TB, 0, stream>>>(bufB, bufA, dinv, b2, N);

    // --- layer 3: 32 -> 2 + log_softmax ---
    k_gemm3<<<cdiv(N, TB), TB, 0, stream>>>(bufB, W3, xw3, N);
    k_zero<<<cdiv((long long)N * 2, TB), TB, 0, stream>>>(agg3, N * 2);
    k_scatter_f2<<<cdiv(E, TB), TB, 0, stream>>>(xw3, esrc, edst, dinv, agg3, E);
    k_logsoftmax2<<<cdiv(N, TB), TB, 0, stream>>>(agg3, xw3, dinv, b3, (float*)d_out, N);
}